// SparseGCNLayer_37203006718668
// MI455X (gfx1250) — compile-verified
//
#include <hip/hip_runtime.h>
#include <hip/hip_bf16.h>
#include <math.h>

typedef float v2f __attribute__((ext_vector_type(2)));
typedef float v8f __attribute__((ext_vector_type(8)));

#define D 128            // D_IN == D_OUT
#define KORD 3           // K+1 taps

// ---------------------------------------------------------------- zero init
__global__ void gcn_zero_f4(float4* __restrict__ p, long n4) {
    long i = (long)blockIdx.x * blockDim.x + threadIdx.x;
    if (i < n4) p[i] = make_float4(0.f, 0.f, 0.f, 0.f);
}

// ---------------------------------------------------------------- degree
__global__ void gcn_deg_accum(const float* __restrict__ vals,
                              const int* __restrict__ rows,
                              float* __restrict__ deg, int E) {
    int e = blockIdx.x * blockDim.x + threadIdx.x;
    if (e < E) unsafeAtomicAdd(&deg[rows[e]], vals[e]);
}

// ---------------------------------------------------------------- COO SpMM
// one wave (32 lanes x float4) per edge; 8 edges per 256-thread block
__global__ void gcn_spmm(const float* __restrict__ hin,
                         const float* __restrict__ vals,
                         const int* __restrict__ rows,
                         const int* __restrict__ cols,
                         float* __restrict__ hout, int E) {
    int e = blockIdx.x * 8 + threadIdx.y;
    if (e >= E) return;
    const int r = rows[e];
    const int c = cols[e];
    const float v = vals[e];
    const float4 d = ((const float4*)(hin + (size_t)c * D))[threadIdx.x];
    float* dst = hout + (size_t)r * D + threadIdx.x * 4;
    unsafeAtomicAdd(dst + 0, v * d.x);
    unsafeAtomicAdd(dst + 1, v * d.y);
    unsafeAtomicAdd(dst + 2, v * d.z);
    unsafeAtomicAdd(dst + 3, v * d.w);
}

// ---------------------------------------------------------------- row scale
// y = deg_inv[n] * h[n,:], non-finite masked to zero (matches reference)
__global__ void gcn_scale_norm(float* __restrict__ h,
                               const float* __restrict__ deg, int N) {
    int idx = blockIdx.x * blockDim.x + threadIdx.x;   // one float4 each
    int n = idx >> 5;                                  // 32 float4 per row
    if (n >= N) return;
    float di = 1.0f / deg[n];
    if (!isfinite(di)) di = 0.0f;
    float4* p = (float4*)h + idx;
    float4 y = *p;
    y.x *= di; y.y *= di; y.z *= di; y.w *= di;
    if (!isfinite(y.x)) y.x = 0.0f;
    if (!isfinite(y.y)) y.y = 0.0f;
    if (!isfinite(y.z)) y.z = 0.0f;
    if (!isfinite(y.w)) y.w = 0.0f;
    *p = y;
}

// ---------------------------------------------------------------- final GEMM
// out[n,o] = sum_k sum_i F_k[n,i] * W[i,o,k]  via V_WMMA_F32_16X16X4_F32.
// Block: 256 threads = 8 wave32; wave w computes the 16x16 tile
// (rows m0..m0+15, cols 16w..16w+15). EXEC all-ones throughout (WMMA req).
__global__ void gcn_gemm_wmma(const float* __restrict__ x,
                              const float* __restrict__ h1,
                              const float* __restrict__ h2,
                              const float* __restrict__ W,   // [128,128,3]
                              float* __restrict__ out) {
    const int wave = threadIdx.x >> 5;     // 0..7 -> output-feature tile
    const int lane = threadIdx.x & 31;
    const int m0   = blockIdx.x * 16;
    const int n0   = wave * 16;
    const int r    = lane & 15;            // row (A) / col (B) within tile
    const int koff = (lane >> 4) * 2;      // K slots {0,1} or {2,3}

    const float* feats[3] = { x, h1, h2 };

    v8f acc = {};
    #pragma unroll
    for (int k = 0; k < KORD; ++k) {
        const float* Arow = feats[k] + (size_t)(m0 + r) * D;
        const float* Wcol = W + (size_t)(n0 + r) * KORD + k;  // stride 384 over i
        for (int i0 = 0; i0 < D; i0 += 4) {
            v2f a, b;
            a.x = Arow[i0 + koff];
            a.y = Arow[i0 + koff + 1];
            b.x = Wcol[(size_t)(i0 + koff)     * (D * KORD)];
            b.y = Wcol[(size_t)(i0 + koff + 1) * (D * KORD)];
            acc = __builtin_amdgcn_wmma_f32_16x16x4_f32(
                false, a, false, b, (short)0, acc, false, false);
        }
    }

    // C/D layout: VGPR v, lanes 0-15 -> (M=v, N=lane); lanes 16-31 -> (M=v+8)
    float* o = out + (size_t)(m0 + 8 * (lane >> 4)) * D + n0 + r;
    #pragma unroll
    for (int v = 0; v < 8; ++v) o[(size_t)v * D] = acc[v];
}

// ---------------------------------------------------------------- launcher
extern "C" void kernel_launch(void* const* d_in, const int* in_sizes, int n_in,
                              void* d_out, int out_size, void* d_ws, size_t ws_size,
                              hipStream_t stream) {
    const float* x    = (const float*)d_in[0];   // [N,128]
    const float* vals = (const float*)d_in[1];   // [E]
    const float* W    = (const float*)d_in[2];   // [128,128,3]
    const int*   rows = (const int*)d_in[3];     // [E]
    const int*   cols = (const int*)d_in[4];     // [E]
    float*       out  = (float*)d_out;           // [N,128]

    const int N = in_sizes[0] / D;               // 100000
    const int E = in_sizes[1];                   // 1600000

    float* h1  = (float*)d_ws;
    float* h2  = h1 + (size_t)N * D;
    float* deg = h2 + (size_t)N * D;

    // zero h1, h2, deg (contiguous: N*(2*128+1) floats, divisible by 4)
    {
        long n4 = ((long)N * (2 * D + 1)) / 4;
        long blocks = (n4 + 255) / 256;
        gcn_zero_f4<<<(int)blocks, 256, 0, stream>>>((float4*)d_ws, n4);
    }

    gcn_deg_accum<<<(E + 255) / 256, 256, 0, stream>>>(vals, rows, deg, E);

    dim3 spmmBlk(32, 8);
    int spmmGrid = (E + 7) / 8;
    int scaleGrid = (N * (D / 4) + 255) / 256;

    gcn_spmm<<<spmmGrid, spmmBlk, 0, stream>>>(x, vals, rows, cols, h1, E);
    gcn_scale_norm<<<scaleGrid, 256, 0, stream>>>(h1, deg, N);

    gcn_spmm<<<spmmGrid, spmmBlk, 0, stream>>>(h1, vals, rows, cols, h2, E);
    gcn_scale_norm<<<scaleGrid, 256, 0, stream>>>(h2, deg, N);

    gcn_gemm_wmma<<<N / 16, 256, 0, stream>>>(x, h1, h2, W, out);
}